// LossPerId_4698694221868
// MI455X (gfx1250) — compile-verified
//
#include <hip/hip_runtime.h>

#define NUM_GROUPS 256
#define NUM_CLASSES 5

// clang ext_vector types: same memory layout as float4/int4 but accepted by
// __builtin_nontemporal_load and guaranteed to lower to one global_load_b128
// with th:TH_LOAD_NT (read-once streaming data -> non-temporal policy).
typedef float vfloat4 __attribute__((ext_vector_type(4)));
typedef int   vint4   __attribute__((ext_vector_type(4)));

// ---------------------------------------------------------------------------
// Kernel 1: zero the global accumulators in d_ws (harness poisons with 0xAA).
// ---------------------------------------------------------------------------
__global__ void lpi_zero_ws(float* __restrict__ ws) {
    int t = blockIdx.x * blockDim.x + threadIdx.x;
    if (t < 2 * NUM_GROUPS) ws[t] = 0.0f;
}

// ---------------------------------------------------------------------------
// Kernel 2: streaming CE + segmented weighted reduction.
//   - 4 samples per thread per grid-stride step: y_pred rows for samples
//     4c..4c+3 are 20 floats = 80B at a 16B-aligned base -> 5x b128 NT loads.
//   - LDS bins (256 num + 256 den) via ds_add_f32, flushed once per block
//     with global_atomic_add_f32.
// ---------------------------------------------------------------------------
__global__ __launch_bounds__(256)
void lpi_main(const float* __restrict__ y_pred,
              const int*   __restrict__ y_true,
              const int*   __restrict__ cluster_ids,
              const float* __restrict__ weights,
              float*       __restrict__ ws,
              int n_chunks) {
    __shared__ float s_num[NUM_GROUPS];
    __shared__ float s_den[NUM_GROUPS];

    for (int t = threadIdx.x; t < NUM_GROUPS; t += blockDim.x) {
        s_num[t] = 0.0f;
        s_den[t] = 0.0f;
    }
    __syncthreads();

    const int stride = gridDim.x * blockDim.x;
    for (int c = blockIdx.x * blockDim.x + threadIdx.x; c < n_chunks; c += stride) {
        // Prefetch next grid-stride chunk of the fat stream (y_pred) into GL2.
        // Speculative: OOB prefetches are silently dropped by hardware.
        __builtin_prefetch(&y_pred[(size_t)(c + stride) * (4 * NUM_CLASSES)], 0, 1);

        // ---- fully vectorized non-temporal loads: 8x global_load_b128 ----
        const vfloat4* yp4 = (const vfloat4*)(y_pred + (size_t)c * (4 * NUM_CLASSES));
        vfloat4 q[5];
#pragma unroll
        for (int i = 0; i < 5; ++i) q[i] = __builtin_nontemporal_load(&yp4[i]);

        float r[4 * NUM_CLASSES];
#pragma unroll
        for (int i = 0; i < 5; ++i) {
#pragma unroll
            for (int j = 0; j < 4; ++j) r[4 * i + j] = q[i][j];
        }

        const vfloat4 w4 = __builtin_nontemporal_load((const vfloat4*)weights + c);
        const vint4   g4 = __builtin_nontemporal_load((const vint4*)cluster_ids + c);
        const vint4   l4 = __builtin_nontemporal_load((const vint4*)y_true + c);

#pragma unroll
        for (int s = 0; s < 4; ++s) {
            const float x0 = r[5 * s + 0];
            const float x1 = r[5 * s + 1];
            const float x2 = r[5 * s + 2];
            const float x3 = r[5 * s + 3];
            const float x4 = r[5 * s + 4];

            const float m = fmaxf(fmaxf(fmaxf(x0, x1), fmaxf(x2, x3)), x4);
            const float sum = __expf(x0 - m) + __expf(x1 - m) + __expf(x2 - m)
                            + __expf(x3 - m) + __expf(x4 - m);
            const float lse = m + __logf(sum);

            const int l = l4[s];
            const float xl = (l == 0) ? x0
                           : (l == 1) ? x1
                           : (l == 2) ? x2
                           : (l == 3) ? x3 : x4;

            const float nll = lse - xl;
            const float wt  = w4[s];
            const int   g   = g4[s];

            // ds_add_f32 (no-return LDS float atomic), DScnt tracked
            atomicAdd(&s_num[g], wt * nll);
            atomicAdd(&s_den[g], wt);
        }
    }
    __syncthreads();

    // Flush block partials: hardware global_atomic_add_f32 (no CAS loop).
    for (int t = threadIdx.x; t < NUM_GROUPS; t += blockDim.x) {
        unsafeAtomicAdd(&ws[t], s_num[t]);
        unsafeAtomicAdd(&ws[NUM_GROUPS + t], s_den[t]);
    }
}

// ---------------------------------------------------------------------------
// Kernel 3: out[g] = num[g] / den[g]
// ---------------------------------------------------------------------------
__global__ void lpi_finalize(const float* __restrict__ ws, float* __restrict__ out) {
    int t = blockIdx.x * blockDim.x + threadIdx.x;
    if (t < NUM_GROUPS) out[t] = ws[t] / ws[NUM_GROUPS + t];
}

// ---------------------------------------------------------------------------
// Host entry
// ---------------------------------------------------------------------------
extern "C" void kernel_launch(void* const* d_in, const int* in_sizes, int n_in,
                              void* d_out, int out_size, void* d_ws, size_t ws_size,
                              hipStream_t stream) {
    const float* y_pred      = (const float*)d_in[0];
    const int*   y_true      = (const int*)  d_in[1];
    const int*   cluster_ids = (const int*)  d_in[2];
    const float* weights     = (const float*)d_in[3];
    float* out = (float*)d_out;
    float* ws  = (float*)d_ws;   // layout: [256 num][256 den]

    const int n        = in_sizes[1];   // N samples (y_true length)
    const int n_chunks = n / 4;         // N = 2^23, divisible by 4

    lpi_zero_ws<<<2, 256, 0, stream>>>(ws);

    const int blocks  = 512;            // 4096 wave32s; 16 iters/thread at N=8M
    const int threads = 256;            // 8 waves/block on CDNA5
    lpi_main<<<blocks, threads, 0, stream>>>(y_pred, y_true, cluster_ids,
                                             weights, ws, n_chunks);

    lpi_finalize<<<1, 256, 0, stream>>>(ws, out);
}